// GCN_36945308680385
// MI455X (gfx1250) — compile-verified
//
#include <hip/hip_runtime.h>
#include <hip/hip_bf16.h>

// Problem sizes (fixed by the reference)
#define Nn 100000
#define Ee 1600000
#define Ff 64
#define FEe 16
#define Hh 128
#define Gg 512
#define HF (Hh + FEe)   // 144

typedef __attribute__((ext_vector_type(2))) float v2f;
typedef __attribute__((ext_vector_type(8))) float v8f;

// ---------------------------------------------------------------------------
// Degree / graph-count / dinv
// ---------------------------------------------------------------------------
__global__ void deg_kernel(const int* __restrict__ dst, float* __restrict__ deg, int e) {
    int i = blockIdx.x * blockDim.x + threadIdx.x;
    if (i < e) atomicAdd(&deg[dst[i]], 1.0f);
}

__global__ void cnt_kernel(const int* __restrict__ batch, float* __restrict__ cnt, int n) {
    int i = blockIdx.x * blockDim.x + threadIdx.x;
    if (i < n) atomicAdd(&cnt[batch[i]], 1.0f);
}

__global__ void dinv_kernel(float* __restrict__ deg, int n) {
    int i = blockIdx.x * blockDim.x + threadIdx.x;
    if (i < n) deg[i] = rsqrtf(deg[i] + 1.0f);   // in place: deg -> dinv
}

// ---------------------------------------------------------------------------
// WMMA GEMM: Y[nRows x 128] = X[nRows x K] * W[K x 128]
// One wave computes a 16-row M-tile across all 8 N-tiles (reuses A fragment).
// W staged in LDS with stride 132 (bank-conflict-free for the +2 K lane split).
// ---------------------------------------------------------------------------
__global__ __launch_bounds__(256) void gemm_wmma_kernel(
    const float* __restrict__ X, const float* __restrict__ W,
    float* __restrict__ Y, int nRows, int K) {
    extern __shared__ float lds[];   // K * 132 floats

    // Cooperative load of W into LDS (all threads, before any early-out)
    const int nv4 = (K * Hh) >> 2;
    for (int idx = threadIdx.x; idx < nv4; idx += blockDim.x) {
        int lin = idx << 2;
        int r = lin >> 7;           // row    (0..K-1)
        int c = lin & 127;          // column (0..127), multiple of 4
        float4 v = *(const float4*)(W + lin);
        *(float4*)(&lds[r * 132 + c]) = v;
    }
    __syncthreads();

    const int wave = threadIdx.x >> 5;
    const int lane = threadIdx.x & 31;
    const int mTile = blockIdx.x * 8 + wave;
    if (mTile * 16 >= nRows) return;            // wave-uniform exit: EXEC stays all-1s

    const int mBase   = mTile * 16;
    const int laneLow = lane & 15;
    const int koff    = (lane >> 4) << 1;       // lanes 16..31 carry K+2 / K+3
    const float* xrow = X + (size_t)(mBase + laneLow) * K;

    v8f acc[8];
    const v8f vzero = {0.f, 0.f, 0.f, 0.f, 0.f, 0.f, 0.f, 0.f};
#pragma unroll
    for (int t = 0; t < 8; ++t) acc[t] = vzero;

    for (int k = 0; k < K; k += 4) {
        v2f a = *(const v2f*)(xrow + k + koff); // A: 16x4 fragment (2 VGPRs)
        const float* w0 = lds + (k + koff) * 132;
#pragma unroll
        for (int t = 0; t < 8; ++t) {
            int col = t * 16 + laneLow;
            v2f b;
            b.x = w0[col];                       // B row k+koff
            b.y = w0[col + 132];                 // B row k+koff+1
            acc[t] = __builtin_amdgcn_wmma_f32_16x16x4_f32(
                false, a, false, b, (short)0, acc[t], false, false);
        }
    }

    // Store: VGPR i -> row mBase + i (+8 for lanes 16..31), col = nBase + laneLow
    const int mOff = (lane >> 4) * 8;
#pragma unroll
    for (int t = 0; t < 8; ++t) {
        float* yp = Y + (size_t)mBase * Hh + t * 16 + laneLow;
#pragma unroll
        for (int i = 0; i < 8; ++i) {
            yp[(size_t)(i + mOff) * Hh] = acc[t][i];
        }
    }
}

// ---------------------------------------------------------------------------
// Edge scatter: agg[dst] += hpre[src] * dinv[src]*dinv[dst]   (32 lanes/edge)
// ---------------------------------------------------------------------------
__global__ __launch_bounds__(256) void scatter_h_kernel(
    const float* __restrict__ hpre, const int* __restrict__ src,
    const int* __restrict__ dst, const float* __restrict__ dinv,
    float* __restrict__ agg, int e) {
    int tid = blockIdx.x * blockDim.x + threadIdx.x;
    int ed = tid >> 5;
    if (ed >= e) return;
    int lane = tid & 31;
    int s = src[ed], d = dst[ed];
    float w = dinv[s] * dinv[d];
    float4 v = *(const float4*)(hpre + (size_t)s * Hh + lane * 4);
    float* p = agg + (size_t)d * Hh + lane * 4;
    atomicAdd(p + 0, v.x * w);
    atomicAdd(p + 1, v.y * w);
    atomicAdd(p + 2, v.z * w);
    atomicAdd(p + 3, v.w * w);
}

// ---------------------------------------------------------------------------
// Self-loop term + bias (+ optional ReLU), in place on agg
// ---------------------------------------------------------------------------
__global__ void finish_kernel(float* __restrict__ agg, const float* __restrict__ hpre,
                              const float* __restrict__ dinv, const float* __restrict__ bias,
                              int n, int do_relu) {
    int tid = blockIdx.x * blockDim.x + threadIdx.x;
    if (tid >= n * Hh) return;
    int node = tid >> 7;
    int j = tid & 127;
    float dv = dinv[node];
    float v = agg[tid] + hpre[tid] * dv * dv + bias[j];
    if (do_relu) v = fmaxf(v, 0.0f);
    agg[tid] = v;
}

// ---------------------------------------------------------------------------
// Edge-attr scatter to both endpoints (4 lanes/edge, float4 each)
// ---------------------------------------------------------------------------
__global__ void eagg_kernel(const float* __restrict__ edge_attr, const int* __restrict__ src,
                            const int* __restrict__ dst, float* __restrict__ eagg, int e) {
    int tid = blockIdx.x * blockDim.x + threadIdx.x;
    if (tid >= e * 4) return;
    int ed = tid >> 2;
    int q = (tid & 3) * 4;
    int s = src[ed], d = dst[ed];
    float4 v = *(const float4*)(edge_attr + (size_t)ed * FEe + q);
    float* ps = eagg + (size_t)s * FEe + q;
    float* pd = eagg + (size_t)d * FEe + q;
    atomicAdd(ps + 0, v.x); atomicAdd(ps + 1, v.y);
    atomicAdd(ps + 2, v.z); atomicAdd(ps + 3, v.w);
    atomicAdd(pd + 0, v.x); atomicAdd(pd + 1, v.y);
    atomicAdd(pd + 2, v.z); atomicAdd(pd + 3, v.w);
}

// ---------------------------------------------------------------------------
// Graph pooling: sums[batch[n]][0:144] += concat(h2[n], eagg[n])
// ---------------------------------------------------------------------------
__global__ void pool_kernel(const float* __restrict__ h, const float* __restrict__ eagg,
                            const int* __restrict__ batch, float* __restrict__ sums, int n) {
    int tid = blockIdx.x * blockDim.x + threadIdx.x;
    if (tid >= n * (HF / 4)) return;
    int node = tid / (HF / 4);
    int q = tid - node * (HF / 4);
    int f = q * 4;
    int g = batch[node];
    float4 v;
    if (f < Hh) v = *(const float4*)(h + (size_t)node * Hh + f);
    else        v = *(const float4*)(eagg + (size_t)node * FEe + (f - Hh));
    float* p = sums + (size_t)g * HF + f;
    atomicAdd(p + 0, v.x); atomicAdd(p + 1, v.y);
    atomicAdd(p + 2, v.z); atomicAdd(p + 3, v.w);
}

// ---------------------------------------------------------------------------
// Final head: out[g] = (sums[g] . Wl) / max(cnt[g],1) + bl
// ---------------------------------------------------------------------------
__global__ void final_kernel(const float* __restrict__ sums, const float* __restrict__ cnt,
                             const float* __restrict__ Wl, const float* __restrict__ bl,
                             float* __restrict__ out, int g) {
    int i = blockIdx.x * blockDim.x + threadIdx.x;
    if (i >= g) return;
    const float* row = sums + (size_t)i * HF;
    float s = 0.0f;
    for (int j = 0; j < HF; ++j) s += row[j] * Wl[j];
    out[i] = s / fmaxf(cnt[i], 1.0f) + bl[0];
}

// ---------------------------------------------------------------------------
extern "C" void kernel_launch(void* const* d_in, const int* in_sizes, int n_in,
                              void* d_out, int out_size, void* d_ws, size_t ws_size,
                              hipStream_t stream) {
    const float* x         = (const float*)d_in[0];
    const int*   edge_idx  = (const int*)d_in[1];   // [2,E]: row0=src, row1=dst
    const float* edge_attr = (const float*)d_in[2];
    const int*   batch     = (const int*)d_in[3];
    const float* W1 = (const float*)d_in[4];
    const float* b1 = (const float*)d_in[5];
    const float* W2 = (const float*)d_in[6];
    const float* b2 = (const float*)d_in[7];
    const float* Wl = (const float*)d_in[8];
    const float* bl = (const float*)d_in[9];
    float* out = (float*)d_out;

    const int* src = edge_idx;
    const int* dst = edge_idx + Ee;

    // Workspace partition (floats)
    float* ws   = (float*)d_ws;
    float* bufA = ws;                               // [N,H]  hpre
    float* bufB = bufA + (size_t)Nn * Hh;           // [N,H]  agg / h
    float* dinv = bufB + (size_t)Nn * Hh;           // [N]    deg -> dinv
    float* cnt  = dinv + Nn;                        // [G]
    float* eagg = cnt + Gg;                         // [N,FE]
    float* sums = eagg + (size_t)Nn * FEe;          // [G,144]

    // Zero accumulators (capture-safe async memsets)
    hipMemsetAsync(bufB, 0, (size_t)Nn * Hh * sizeof(float), stream);
    hipMemsetAsync(dinv, 0, (size_t)(Nn + Gg) * sizeof(float), stream);
    hipMemsetAsync(eagg, 0, (size_t)Nn * FEe * sizeof(float), stream);
    hipMemsetAsync(sums, 0, (size_t)Gg * HF * sizeof(float), stream);

    const int T = 256;
    // Degrees / counts / dinv
    deg_kernel<<<(Ee + T - 1) / T, T, 0, stream>>>(dst, dinv, Ee);
    cnt_kernel<<<(Nn + T - 1) / T, T, 0, stream>>>(batch, cnt, Nn);
    dinv_kernel<<<(Nn + T - 1) / T, T, 0, stream>>>(dinv, Nn);

    const int mTiles = Nn / 16;                     // 6250 (N % 16 == 0)
    const int gemmBlocks = (mTiles + 7) / 8;

    // ----- Layer 1: hpre = x @ W1 -----
    gemm_wmma_kernel<<<gemmBlocks, 256, (size_t)Ff * 132 * sizeof(float), stream>>>(
        x, W1, bufA, Nn, Ff);
    scatter_h_kernel<<<(Ee * 32 + T - 1) / T, T, 0, stream>>>(bufA, src, dst, dinv, bufB, Ee);
    finish_kernel<<<(Nn * Hh + T - 1) / T, T, 0, stream>>>(bufB, bufA, dinv, b1, Nn, 1);

    // ----- Layer 2: hpre2 = h1 @ W2 -----
    gemm_wmma_kernel<<<gemmBlocks, 256, (size_t)Hh * 132 * sizeof(float), stream>>>(
        bufB, W2, bufA, Nn, Hh);
    hipMemsetAsync(bufB, 0, (size_t)Nn * Hh * sizeof(float), stream);
    scatter_h_kernel<<<(Ee * 32 + T - 1) / T, T, 0, stream>>>(bufA, src, dst, dinv, bufB, Ee);
    finish_kernel<<<(Nn * Hh + T - 1) / T, T, 0, stream>>>(bufB, bufA, dinv, b2, Nn, 0);

    // ----- Edge features + pooling + head -----
    eagg_kernel<<<(Ee * 4 + T - 1) / T, T, 0, stream>>>(edge_attr, src, dst, eagg, Ee);
    pool_kernel<<<(Nn * (HF / 4) + T - 1) / T, T, 0, stream>>>(bufB, eagg, batch, sums, Nn);
    final_kernel<<<(Gg + T - 1) / T, T, 0, stream>>>(sums, cnt, Wl, bl, out, Gg);
}